// GroupScore_88502096101482
// MI455X (gfx1250) — compile-verified
//
#include <hip/hip_runtime.h>
#include <stdint.h>

#define BB    32
#define NN    262144
#define GG    256
#define CH    16                 // chunks per batch (phase1 blocks per batch)
#define CHUNK (NN / CH)          // 16384 elements per phase1 block
#define TS    2048               // TDM tile size (elements)
#define TILES (CHUNK / TS)       // 8
#define BPB   64                 // phase3 blocks per batch

typedef uint32_t u32x4 __attribute__((ext_vector_type(4)));
typedef uint32_t u32x8 __attribute__((ext_vector_type(8)));

#if __has_builtin(__builtin_amdgcn_s_wait_tensorcnt)
#define WAIT_TCNT(n) __builtin_amdgcn_s_wait_tensorcnt(n)
#else
#define WAIT_TCNT(n) asm volatile("s_wait_tensorcnt %0" ::"i"(n) : "memory")
#endif

// ---- order-preserving float<->u32 key (for ds_min_u32 / ds_max_u32) ----
__device__ __forceinline__ uint32_t fkey(float x) {
  uint32_t u = __float_as_uint(x);
  return (u & 0x80000000u) ? ~u : (u | 0x80000000u);
}
__device__ __forceinline__ float funkey(uint32_t k) {
  return __uint_as_float((k & 0x80000000u) ? (k ^ 0x80000000u) : ~k);
}

// ---- Tensor Data Mover: 1-D tile, 4-byte elements, Global -> LDS ----------
// D# layout per CDNA5 ISA (group0 128b, group1 256b).
// Inline asm (not the clang builtin) so it assembles on both toolchains.
__device__ __forceinline__ void tdm_load_1d(uint32_t lds_addr, const void* gptr,
                                            uint32_t n_elems) {
  uint64_t ga = (uint64_t)(uintptr_t)gptr;
  u32x4 g0;
  g0[0] = 0x1u;                                             // count=1, user desc
  g0[1] = lds_addr;                                         // LDS byte address
  g0[2] = (uint32_t)ga;                                     // global_addr[31:0]
  g0[3] = ((uint32_t)(ga >> 32) & 0x01FFFFFFu) | (2u << 30);// addr[56:32], type=2
  u32x8 g1;
  g1[0] = (2u << 16);                      // wg_mask=0, data_size=2 (4 bytes)
  g1[1] = (n_elems & 0xFFFFu) << 16;       // tensor_dim0[15:0]
  g1[2] = (n_elems >> 16) | (1u << 16);    // tensor_dim0[31:16] | tensor_dim1=1
  g1[3] = (n_elems & 0xFFFFu) << 16;       // tile_dim0
  g1[4] = 0u;                              // tile_dim1=0, tile_dim2=0 (unused)
  g1[5] = n_elems;                         // tensor_dim0_stride[31:0]
  g1[6] = 0u;
  g1[7] = 0u;
  asm volatile("tensor_load_to_lds %0, %1" ::"s"(g0), "s"(g1) : "memory");
}

__device__ __forceinline__ uint32_t lds_addr_of(const void* p) {
  return (uint32_t)(uintptr_t)p;  // generic shared ptr: low 32 bits = LDS offset
}

// ============================ Phase 1 ======================================
// Per (batch, chunk): TDM-stage tiles of pr/vr into LDS (double-buffered),
// segmented min/max/sum into LDS tables via DS atomics, write chunk partials.
__global__ __launch_bounds__(256) void phase1_seg_reduce(
    const float* __restrict__ pr, const int* __restrict__ vr,
    float* __restrict__ partial) {
  __shared__ __align__(16) float sP[2][TS];
  __shared__ __align__(16) int   sV[2][TS];
  __shared__ uint32_t skMin[GG];
  __shared__ uint32_t skMax[GG];
  __shared__ float    sSum[GG];

  const int tid = threadIdx.x;
  const int b = blockIdx.x / CH, c = blockIdx.x % CH;
  const size_t base = (size_t)b * NN + (size_t)c * CHUNK;

  skMin[tid] = 0xFF800000u;  // fkey(+inf)
  skMax[tid] = 0x007FFFFFu;  // fkey(-inf)
  sSum[tid]  = 0.0f;
  __syncthreads();

  const bool issuer = (tid < 32);  // wave 0 drives the TDM engine
  if (issuer) {
    tdm_load_1d(lds_addr_of(&sP[0][0]), pr + base, TS);
    tdm_load_1d(lds_addr_of(&sV[0][0]), vr + base, TS);
  }
  for (int t = 0; t < TILES; ++t) {
    const int buf = t & 1;
    if (issuer) {
      if (t + 1 < TILES) {  // prefetch next tile, then wait only on current
        tdm_load_1d(lds_addr_of(&sP[buf ^ 1][0]),
                    pr + base + (size_t)(t + 1) * TS, TS);
        tdm_load_1d(lds_addr_of(&sV[buf ^ 1][0]),
                    vr + base + (size_t)(t + 1) * TS, TS);
        WAIT_TCNT(2);
      } else {
        WAIT_TCNT(0);
      }
    }
    __syncthreads();  // tile t visible to every wave
#pragma unroll
    for (int e = 0; e < TS / 256; e += 4) {
      const int i = tid * (TS / 256) + e;  // 16B-aligned LDS b128 reads
      float4 p4 = *(const float4*)&sP[buf][i];
      int4   v4 = *(const int4*)&sV[buf][i];
      atomicMin(&skMin[v4.x], fkey(p4.x));
      atomicMax(&skMax[v4.x], fkey(p4.x));
      atomicAdd(&sSum[v4.x], p4.x);
      atomicMin(&skMin[v4.y], fkey(p4.y));
      atomicMax(&skMax[v4.y], fkey(p4.y));
      atomicAdd(&sSum[v4.y], p4.y);
      atomicMin(&skMin[v4.z], fkey(p4.z));
      atomicMax(&skMax[v4.z], fkey(p4.z));
      atomicAdd(&sSum[v4.z], p4.z);
      atomicMin(&skMin[v4.w], fkey(p4.w));
      atomicMax(&skMax[v4.w], fkey(p4.w));
      atomicAdd(&sSum[v4.w], p4.w);
    }
    __syncthreads();  // buffer consumed; safe for TDM to overwrite
  }
  float* outp = partial + (size_t)(b * CH + c) * 3 * GG;
  outp[0 * GG + tid] = funkey(skMin[tid]);
  outp[1 * GG + tid] = funkey(skMax[tid]);
  outp[2 * GG + tid] = sSum[tid];
}

// ============================ Phase 2 ======================================
// One block per batch: merge chunk partials, blend means, bitonic argsort of
// 256 keys (index tie-break => stable order), conv + inverse-permute, emit
// per-group affine (a,b) with tmp = a*pr + b.
__global__ __launch_bounds__(256) void phase2_groups(
    const float* __restrict__ partial, const float* __restrict__ inp_means,
    const float* __restrict__ W, float2* __restrict__ tab) {
  __shared__ float sv[GG];
  __shared__ int   si[GG];
  __shared__ float sf0[GG];
  __shared__ float sf1[GG];
  const int g = threadIdx.x, b = blockIdx.x;

  float vmin = __uint_as_float(0x7F800000u);   // +inf
  float vmax = __uint_as_float(0xFF800000u);   // -inf
  float vsum = 0.0f;
  const float* p = partial + (size_t)b * CH * 3 * GG;
  for (int c = 0; c < CH; ++c) {
    vmin = fminf(vmin, p[(c * 3 + 0) * GG + g]);
    vmax = fmaxf(vmax, p[(c * 3 + 1) * GG + g]);
    vsum += p[(c * 3 + 2) * GG + g];
  }
  const bool any = (vmin < __uint_as_float(0x7F800000u));
  const float W0 = W[0], W1 = W[1];
  const float vmean = (inp_means[b * GG + g] * W0 + (any ? vsum : 0.0f) * W1) /
                      (W0 + W1);
  sv[g] = vmean;
  si[g] = g;
  __syncthreads();

  for (int k = 2; k <= GG; k <<= 1) {
    for (int j = k >> 1; j > 0; j >>= 1) {
      const int q = g ^ j;
      if (q > g) {
        float a = sv[g], c2 = sv[q];
        int ia = si[g], ib = si[q];
        const bool up = ((g & k) == 0);
        const bool sw = up ? ((a > c2) || (a == c2 && ia > ib))
                           : ((a < c2) || (a == c2 && ia < ib));
        if (sw) { sv[g] = c2; sv[q] = a; si[g] = ib; si[q] = ia; }
      }
      __syncthreads();
    }
  }
  const float vs   = sv[g];
  const float prev = (g == 0)      ? sv[0]           : sv[g - 1];
  const float nxt  = (g == GG - 1) ? sv[GG - 1] * 2.0f : sv[g + 1];
  const int   og   = si[g];
  sf0[og] = (prev + vs) / 1.999f;
  sf1[og] = (vs + nxt) / 2.001f;
  __syncthreads();

  const bool ns = (vmin == vmax);
  const float f0 = ns ? 0.0f : sf0[g];
  const float f1 = ns ? 1.0f : sf1[g];
  const float lo = ns ? 0.0f : vmin;
  const float hi = ns ? 1.0f : vmax;
  const float a  = (f1 - f0) / (hi - lo);
  const float bb = f0 - lo * a;
  tab[b * GG + g] = make_float2(a, bb);  // empty groups: never gathered
}

// ============================ Phase 3 ======================================
__device__ __forceinline__ float scale_one(float p, float2 ab) {
  const float tmp = fmaf(p, ab.x, ab.y);
  const bool bad = (tmp != tmp) || (tmp == 0.0f);
  const float s = p / (bad ? 1.0f : tmp);
  const bool b2 = bad || (s != s) || (fabsf(s) == __uint_as_float(0x7F800000u));
  return p * (b2 ? 0.0f : s);
}

__global__ __launch_bounds__(256) void phase3_apply(
    const float* __restrict__ pr, const int* __restrict__ vr,
    const float2* __restrict__ tab, float* __restrict__ out) {
  __shared__ float2 sAB[GG];
  const int tid = threadIdx.x;
  const int b = blockIdx.x / BPB, blk = blockIdx.x % BPB;
  sAB[tid] = tab[b * GG + tid];
  __syncthreads();

  const size_t base = (size_t)b * NN + (size_t)blk * (NN / BPB);
  const int ITERS = (NN / BPB) / (256 * 4);  // 4
#pragma unroll
  for (int it = 0; it < ITERS; ++it) {
    const size_t idx = base + (size_t)it * 1024 + (size_t)tid * 4;
    if (it + 1 < ITERS) {                     // global_prefetch_b8 hints
      __builtin_prefetch(pr + idx + 1024, 0, 0);
      __builtin_prefetch(vr + idx + 1024, 0, 0);
    }
    float4 p = *(const float4*)(pr + idx);
    int4   v = *(const int4*)(vr + idx);
    float4 o;
    o.x = scale_one(p.x, sAB[v.x]);
    o.y = scale_one(p.y, sAB[v.y]);
    o.z = scale_one(p.z, sAB[v.z]);
    o.w = scale_one(p.w, sAB[v.w]);
    *(float4*)(out + idx) = o;
  }
}

// ============================ Launch =======================================
extern "C" void kernel_launch(void* const* d_in, const int* in_sizes, int n_in,
                              void* d_out, int out_size, void* d_ws,
                              size_t ws_size, hipStream_t stream) {
  const float* pr        = (const float*)d_in[0];
  const float* inp_means = (const float*)d_in[1];
  const int*   vr        = (const int*)d_in[2];
  const float* W         = (const float*)d_in[3];
  float* out = (float*)d_out;

  float*  partial = (float*)d_ws;  // BB*CH*3*GG floats = 1.5 MiB
  float2* tab = (float2*)((char*)d_ws + (size_t)BB * CH * 3 * GG * sizeof(float));

  phase1_seg_reduce<<<BB * CH, 256, 0, stream>>>(pr, vr, partial);
  phase2_groups<<<BB, 256, 0, stream>>>(partial, inp_means, W, tab);
  phase3_apply<<<BB * BPB, 256, 0, stream>>>(pr, vr, tab, out);
}